// PromptProcessor_58351425683655
// MI455X (gfx1250) — compile-verified
//
#include <hip/hip_runtime.h>
#include <stdint.h>

// Problem constants from the reference
#define BATCH   128
#define VOCAB   50257
#define HISTLEN 2048

#define NBINS 2048
#define TILE  4096           // floats staged per async round (1024 lanes * 4)
#define CAP1  1024           // top-p boundary-bin candidate cap
#define CAP2  256            // top-k boundary-bin candidate cap
#define RANGEF 32.0f
#define BINW   (RANGEF / (float)NBINS)
#define INV_BINW ((float)NBINS / RANGEF)
#define NTHREADS 1024

__device__ __forceinline__ unsigned f2ord(float f) {
  unsigned u = __float_as_uint(f);
  return (u & 0x80000000u) ? ~u : (u | 0x80000000u);
}
__device__ __forceinline__ float ord2f(unsigned o) {
  return (o & 0x80000000u) ? __uint_as_float(o & 0x7fffffffu) : __uint_as_float(~o);
}
__device__ __forceinline__ float next_up(float x) {
  unsigned u = __float_as_uint(x);
  if (x > 0.f) u += 1u; else if (x < 0.f) u -= 1u; else u = 1u;
  return __uint_as_float(u);
}

__device__ __forceinline__ float block_max(float v, volatile float* red) {
  for (int o = 16; o; o >>= 1) v = fmaxf(v, __shfl_xor(v, o, 32));
  int lane = threadIdx.x & 31, wid = threadIdx.x >> 5, nw = blockDim.x >> 5;
  if (lane == 0) red[wid] = v;
  __syncthreads();
  if (threadIdx.x == 0) {
    float m = red[0];
    for (int i = 1; i < nw; ++i) m = fmaxf(m, red[i]);
    red[0] = m;
  }
  __syncthreads();
  v = red[0];
  __syncthreads();
  return v;
}

__device__ __forceinline__ float block_sum(float v, volatile float* red) {
  for (int o = 16; o; o >>= 1) v += __shfl_xor(v, o, 32);
  int lane = threadIdx.x & 31, wid = threadIdx.x >> 5, nw = blockDim.x >> 5;
  if (lane == 0) red[wid] = v;
  __syncthreads();
  if (threadIdx.x == 0) {
    float s = 0.f;
    for (int i = 0; i < nw; ++i) s += red[i];
    red[0] = s;
  }
  __syncthreads();
  v = red[0];
  __syncthreads();
  return v;
}

__global__ __launch_bounds__(NTHREADS)
void sample_row_kernel(const float* __restrict__ logits, const int* __restrict__ prev,
                       const float* __restrict__ uin,   const float* __restrict__ toppP,
                       const float* __restrict__ rpP,   const float* __restrict__ tempP,
                       const int* __restrict__ topkP,   int* __restrict__ out_idx,
                       float* __restrict__ out_probs) {
  __shared__ float    s_hist[NBINS];
  __shared__ unsigned s_cnt[NBINS];
  __shared__ float    s_tileA[TILE];
  __shared__ float    s_tileB[TILE];
  __shared__ float    s_cand[CAP1];
  __shared__ float    s_cand2[CAP2];
  __shared__ float    s_redf[32];
  __shared__ int      s_redi[32];
  __shared__ int      sh_k1, sh_k2;
  __shared__ float    sh_cumB, sh_r, sh_vp, sh_vk, sh_V;
  __shared__ unsigned sh_cntB1, sh_cntB2, sh_candn, sh_cand2n, sh_keptcnt, sh_minOrd, sh_keptTotal;

  const int tid = threadIdx.x;
  const int bd  = blockDim.x;
  const int b   = blockIdx.x;

  const float* lg = logits + (size_t)b * VOCAB;
  const float* uu = uin    + (size_t)b * VOCAB;
  const int*   pv = prev   + (size_t)b * HISTLEN;
  float*       L  = out_probs + (size_t)b * VOCAB;   // scratch, later probs

  const float topp = *toppP;
  const float rp   = *rpP;
  const float invT = 1.0f / fmaxf(*tempP, 1e-5f);
  const int   topk = *topkP;

  // ---- phase 0: copy row into scratch ----
  for (int i = tid; i < VOCAB; i += bd) L[i] = lg[i];
  __syncthreads();

  // ---- phase 1: repetition penalty (reads ORIGINAL logits; duplicate writes identical) ----
  for (int h = tid; h < HISTLEN; h += bd) {
    int t = pv[h];
    float x = lg[t];
    L[t] = (x < 0.f) ? x * rp : x / rp;
  }
  __syncthreads();

  // Alignment: rows are only 4-byte aligned for odd b; carve aligned b128 region.
  const uintptr_t baseAddr = (uintptr_t)L;
  const int pre     = (int)(((16u - (unsigned)(baseAddr & 15u)) & 15u) >> 2);
  const int ntiles  = (VOCAB - pre) / TILE;
  const int tileEnd = pre + ntiles * TILE;

  // Per-lane async issue: one global_load_async_to_lds_b128 per lane covers
  // 16 B; a full 1024-thread block stages a 16 KB tile per issue round.
  auto issueTile = [&](int t, const float* buf) {
    const float* gp = L + pre + t * TILE + tid * 4;
    uint32_t lds = (uint32_t)(uintptr_t)buf + (uint32_t)tid * 16u;
    asm volatile("global_load_async_to_lds_b128 %0, %1, off"
                 :: "v"(lds), "v"((uint64_t)(uintptr_t)gp) : "memory");
  };

  // ---- phase 2: row max (double-buffered async global->LDS staging) ----
  float lmax = -INFINITY;
  for (int i = tid; i < pre; i += bd) lmax = fmaxf(lmax, L[i]);
  if (ntiles > 0) issueTile(0, s_tileA);
  for (int t = 0; t < ntiles; ++t) {
    const float* cur = (t & 1) ? s_tileB : s_tileA;
    const float* nxt = (t & 1) ? s_tileA : s_tileB;
    if (t + 1 < ntiles) {
      issueTile(t + 1, nxt);
      asm volatile("s_wait_asynccnt 0x1" ::: "memory");  // tile t done (in-order)
    } else {
      asm volatile("s_wait_asynccnt 0x0" ::: "memory");
    }
    __syncthreads();
    for (int i = tid; i < TILE; i += bd) lmax = fmaxf(lmax, cur[i]);
    __syncthreads();
  }
  for (int i = tileEnd + tid; i < VOCAB; i += bd) lmax = fmaxf(lmax, L[i]);
  const float M = block_max(lmax, s_redf);

  // ---- phase 3: exp/count histogram (double-buffered async staging + LDS atomics) ----
  for (int i = tid; i < NBINS; i += bd) { s_hist[i] = 0.f; s_cnt[i] = 0u; }
  __syncthreads();
  {
    auto binAcc = [&](float v) {
      int bn = (int)((M - v) * INV_BINW);
      bn = bn < 0 ? 0 : (bn >= NBINS ? NBINS - 1 : bn);
      atomicAdd(&s_hist[bn], __expf(v - M));
      atomicAdd(&s_cnt[bn], 1u);
    };
    for (int i = tid; i < pre; i += bd) binAcc(L[i]);
    if (ntiles > 0) issueTile(0, s_tileA);
    for (int t = 0; t < ntiles; ++t) {
      const float* cur = (t & 1) ? s_tileB : s_tileA;
      const float* nxt = (t & 1) ? s_tileA : s_tileB;
      if (t + 1 < ntiles) {
        issueTile(t + 1, nxt);
        asm volatile("s_wait_asynccnt 0x1" ::: "memory");
      } else {
        asm volatile("s_wait_asynccnt 0x0" ::: "memory");
      }
      __syncthreads();
      for (int i = tid; i < TILE; i += bd) binAcc(cur[i]);
      __syncthreads();
    }
    for (int i = tileEnd + tid; i < VOCAB; i += bd) binAcc(L[i]);
  }
  __syncthreads();

  // ---- phase 4: Z (softmax denominator over ALL tokens) ----
  float zpart = 0.f;
  for (int i = tid; i < NBINS; i += bd) zpart += s_hist[i];
  const float Z = block_sum(zpart, s_redf);

  // ---- phase 5: thread-0 scan for boundary bins ----
  if (tid == 0) {
    const float r = topp * Z;
    float cum = 0.f; unsigned ccum = 0u;
    int k1 = NBINS, k2 = -1;
    float cumB = 0.f; unsigned cntB1 = 0u, cntB2 = 0u;
    for (int k = 0; k < NBINS; ++k) {
      float s = s_hist[k]; unsigned c = s_cnt[k];
      if (k1 == NBINS && (cum + s) > r) { k1 = k; cumB = cum; cntB1 = ccum; }
      if (k2 < 0 && (ccum + c) >= (unsigned)topk) { k2 = k; cntB2 = ccum; }
      cum += s; ccum += c;
    }
    sh_k1 = k1; sh_k2 = k2; sh_cumB = cumB; sh_cntB1 = cntB1; sh_cntB2 = cntB2;
    sh_r = r; sh_candn = 0u; sh_cand2n = 0u;
    sh_keptcnt = 0u; sh_minOrd = 0xFFFFFFFFu; sh_vk = -INFINITY;
  }
  __syncthreads();

  // ---- phase 6: collect boundary-bin candidates ----
  {
    const int k1 = sh_k1, k2 = sh_k2;
    for (int i = tid; i < VOCAB; i += bd) {
      float v = L[i];
      int bn = (int)((M - v) * INV_BINW);
      bn = bn < 0 ? 0 : (bn >= NBINS ? NBINS - 1 : bn);
      if (bn == k1) { unsigned p = atomicAdd(&sh_candn, 1u);  if (p < CAP1) s_cand[p]  = v; }
      if (bn == k2) { unsigned p = atomicAdd(&sh_cand2n, 1u); if (p < CAP2) s_cand2[p] = v; }
    }
  }
  __syncthreads();

  // ---- phase 7: exact top-p pivot within boundary bin ----
  {
    int n1 = (int)sh_candn; if (n1 > CAP1) n1 = CAP1;
    float lmin = INFINITY; unsigned lkept = 0u;
    for (int i = tid; i < n1; i += bd) {
      float c = s_cand[i];
      float S = 0.f;
      for (int j = 0; j < n1; ++j) {
        float cj = s_cand[j];
        if (cj >= c) S += __expf(cj - M);
      }
      bool kept = ((sh_cumB + S) <= sh_r) || (c == M);   // rank-0 always kept
      if (kept) { ++lkept; lmin = fminf(lmin, c); }
    }
    if (lkept) { atomicAdd(&sh_keptcnt, lkept); atomicMin(&sh_minOrd, f2ord(lmin)); }
  }
  __syncthreads();
  if (tid == 0) {
    float vp; unsigned ktot;
    if (sh_k1 >= NBINS) { vp = -INFINITY; ktot = VOCAB; }
    else {
      if (sh_minOrd != 0xFFFFFFFFu) vp = ord2f(sh_minOrd);
      else vp = next_up(M - (float)sh_k1 * BINW);       // all of bin k1 removed
      ktot = sh_cntB1 + sh_keptcnt;
    }
    sh_vp = vp; sh_keptTotal = ktot;
  }
  __syncthreads();

  // ---- phase 8: exact top-k pivot (k-th largest value overall) ----
  if (sh_k2 >= 0) {
    int n2 = (int)sh_cand2n; if (n2 > CAP2) n2 = CAP2;
    for (int i = tid; i < n2; i += bd) {
      float c = s_cand2[i];
      unsigned above = 0u;
      for (int j = 0; j < n2; ++j) if (s_cand2[j] > c) ++above;
      if (sh_cntB2 + above + 1u == (unsigned)topk) sh_vk = c;
    }
  }
  __syncthreads();
  if (tid == 0)
    sh_V = (sh_keptTotal >= (unsigned)topk) ? fmaxf(sh_vp, sh_vk) : sh_vp;
  __syncthreads();
  const float Vstar = sh_V;

  // ---- phase 9: masked-softmax normalizer at temperature ----
  float lz = 0.f;
  for (int i = tid; i < VOCAB; i += bd) {
    float v = L[i];
    if (v >= Vstar) lz += __expf((v - M) * invT);
  }
  const float Zp = block_sum(lz, s_redf);
  const float invZ = 1.0f / Zp;

  // ---- phase 10: write probs + Gumbel-ratio argmax ----
  float best = -INFINITY; int bi = 0x7FFFFFFF;
  for (int i = tid; i < VOCAB; i += bd) {
    float v = L[i];
    float p = (v >= Vstar) ? __expf((v - M) * invT) * invZ : 0.f;
    L[i] = p;                                  // overwrite scratch with probs
    float q = -__logf(uu[i]);
    float rat = p / q;
    if (rat > best || (rat == best && i < bi)) { best = rat; bi = i; }
  }
  for (int o = 16; o; o >>= 1) {
    float rr = __shfl_xor(best, o, 32);
    int   ii = __shfl_xor(bi, o, 32);
    if (rr > best || (rr == best && ii < bi)) { best = rr; bi = ii; }
  }
  const int lane = tid & 31, wid = tid >> 5;
  __syncthreads();
  if (lane == 0) { s_redf[wid] = best; s_redi[wid] = bi; }
  __syncthreads();
  if (tid == 0) {
    float bb = s_redf[0]; int bj = s_redi[0];
    const int nw = bd >> 5;
    for (int i = 1; i < nw; ++i)
      if (s_redf[i] > bb || (s_redf[i] == bb && s_redi[i] < bj)) { bb = s_redf[i]; bj = s_redi[i]; }
    out_idx[b] = bj;
  }
}

extern "C" void kernel_launch(void* const* d_in, const int* in_sizes, int n_in,
                              void* d_out, int out_size, void* d_ws, size_t ws_size,
                              hipStream_t stream) {
  (void)in_sizes; (void)n_in; (void)out_size; (void)d_ws; (void)ws_size;
  const float* logits = (const float*)d_in[0];
  const int*   prev   = (const int*)d_in[1];
  const float* u      = (const float*)d_in[2];
  const float* topp   = (const float*)d_in[3];
  const float* rp     = (const float*)d_in[4];
  const float* temp   = (const float*)d_in[5];
  const int*   topk   = (const int*)d_in[6];

  int*   out_idx   = (int*)d_out;                 // first B int32 slots
  float* out_probs = (float*)d_out + BATCH;       // then B*V float probs

  sample_row_kernel<<<BATCH, NTHREADS, 0, stream>>>(
      logits, prev, u, topp, rp, temp, topk, out_idx, out_probs);
}